// EGNN_41369124995426
// MI455X (gfx1250) — compile-verified
//
#include <hip/hip_runtime.h>

// ---------------------------------------------------------------------------
// GCN (8-layer, HID=16) on MI455X / gfx1250, wave32.
// Dense node transform uses V_WMMA_F32_16X16X4_F32; edge aggregation uses
// native f32 global atomics (L2-resident working set: 192MB L2 >> ~130MB).
// ---------------------------------------------------------------------------

typedef __attribute__((ext_vector_type(2))) float v2f;
typedef __attribute__((ext_vector_type(8))) float v8f;

#define HID 16

static __device__ __forceinline__ void atomAddF(float* p, float v) {
    unsafeAtomicAdd(p, v);   // lowers to global_atomic_add_f32 (no CAS loop)
}

// ---------------- degree / norm precompute ----------------

__global__ void k_init_deg(float* __restrict__ deg, int n) {
    int i = blockIdx.x * blockDim.x + threadIdx.x;
    if (i < n) deg[i] = 1.0f;            // self-loop weight 1
}

__global__ void k_deg_acc(const int* __restrict__ col, const float* __restrict__ w,
                          float* __restrict__ deg, int E) {
    int e = blockIdx.x * blockDim.x + threadIdx.x;
    if (e < E) atomAddF(deg + col[e], w[e]);
}

__global__ void k_norm(const int* __restrict__ row, const int* __restrict__ col,
                       const float* __restrict__ w, const float* __restrict__ deg,
                       float* __restrict__ nrm, int E) {
    int e = blockIdx.x * blockDim.x + threadIdx.x;
    if (e >= E) return;
    nrm[e] = rsqrtf(deg[row[e]]) * w[e] * rsqrtf(deg[col[e]]);
}

__global__ void k_deginv(const float* __restrict__ deg, float* __restrict__ dinv, int n) {
    int i = blockIdx.x * blockDim.x + threadIdx.x;
    if (i < n) dinv[i] = 1.0f / deg[i];
}

// ---------------- dense transforms via WMMA ----------------
// One wave handles one 16-node tile: D[16,16] = A[16,K] @ W[K,16].
// A layout (32-bit A, 16x4, ISA 7.12.2): lane m (0..15) holds K=k0,k0+1;
// lane m+16 holds K=k0+2,k0+3 of row m. B row-striped across lanes per VGPR.
// Epilogue writes hw (for the edge gather) and agg = deg_inv*hw + bias.

__global__ __launch_bounds__(256) void k_gemm16(
    const float* __restrict__ hin, const float* __restrict__ W,
    const float* __restrict__ bias, const float* __restrict__ dinv,
    float* __restrict__ hw, float* __restrict__ agg, int ntiles)
{
    int lane = threadIdx.x & 31;
    int wid  = threadIdx.x >> 5;
    int tile = blockIdx.x * 8 + wid;
    if (tile >= ntiles) return;          // wave-uniform: EXEC stays all-ones

    int base = tile * 16;
    int half = lane >> 4;                // 0: K even-pair low, 1: high
    int m    = lane & 15;                // row (A) / column (B,D)
    const float* Arow = hin + (size_t)(base + m) * HID;

    v8f acc = {};
#pragma unroll
    for (int kc = 0; kc < 4; ++kc) {
        int k0 = kc * 4 + 2 * half;
        v2f a = { Arow[k0], Arow[k0 + 1] };
        v2f b = { W[k0 * HID + m], W[(k0 + 1) * HID + m] };
        acc = __builtin_amdgcn_wmma_f32_16x16x4_f32(
            false, a, false, b, (short)0, acc, false, false);
    }

    float bn = bias[m];
#pragma unroll
    for (int v = 0; v < 8; ++v) {
        int M = v + 8 * half;
        size_t idx = (size_t)(base + M) * HID + m;
        float d = acc[v];
        hw[idx]  = d;
        agg[idx] = dinv[base + M] * d + bn;
    }
}

// Layer 1: input is [N,4], so K=4 -> exactly one WMMA per tile.
__global__ __launch_bounds__(256) void k_gemm1(
    const float* __restrict__ x, const float* __restrict__ W1,
    const float* __restrict__ bias, const float* __restrict__ dinv,
    float* __restrict__ hw, float* __restrict__ agg, int ntiles)
{
    int lane = threadIdx.x & 31;
    int wid  = threadIdx.x >> 5;
    int tile = blockIdx.x * 8 + wid;
    if (tile >= ntiles) return;

    int base = tile * 16;
    int half = lane >> 4;
    int m    = lane & 15;
    const float* Arow = x + (size_t)(base + m) * 4;

    int k0 = 2 * half;
    v2f a = { Arow[k0], Arow[k0 + 1] };
    v2f b = { W1[k0 * HID + m], W1[(k0 + 1) * HID + m] };
    v8f acc = {};
    acc = __builtin_amdgcn_wmma_f32_16x16x4_f32(
        false, a, false, b, (short)0, acc, false, false);

    float bn = bias[m];
#pragma unroll
    for (int v = 0; v < 8; ++v) {
        int M = v + 8 * half;
        size_t idx = (size_t)(base + M) * HID + m;
        float d = acc[v];
        hw[idx]  = d;
        agg[idx] = dinv[base + M] * d + bn;
    }
}

// ---------------- edge scatter (dominant cost) ----------------

__global__ __launch_bounds__(256) void k_edge(
    const int* __restrict__ row, const int* __restrict__ col,
    const float* __restrict__ nrm, const float* __restrict__ hw,
    float* __restrict__ agg, int E)
{
    int e = blockIdx.x * blockDim.x + threadIdx.x;
    if (e >= E) return;
    int r = row[e], c = col[e];
    float w = nrm[e];
    const float4* src = (const float4*)(hw + (size_t)r * HID);
    float* dst = agg + (size_t)c * HID;
#pragma unroll
    for (int q = 0; q < 4; ++q) {
        float4 v = src[q];
        atomAddF(dst + q * 4 + 0, w * v.x);
        atomAddF(dst + q * 4 + 1, w * v.y);
        atomAddF(dst + q * 4 + 2, w * v.z);
        atomAddF(dst + q * 4 + 3, w * v.w);
    }
}

// ---------------- activation / residual ----------------
// mode 0: h = relu(agg)
// mode 1: h = relu(h + agg)
// mode 2: h = relu(h + relu(agg))
__global__ void k_post(float* __restrict__ h, const float* __restrict__ agg,
                       int n, int mode) {
    int i = blockIdx.x * blockDim.x + threadIdx.x;
    if (i >= n) return;
    float a = agg[i];
    if (mode == 0)      h[i] = fmaxf(a, 0.0f);
    else if (mode == 1) h[i] = fmaxf(h[i] + a, 0.0f);
    else                h[i] = fmaxf(h[i] + fmaxf(a, 0.0f), 0.0f);
}

// ---------------- pooling + head ----------------

__global__ void k_zero(float* __restrict__ p, int n) {
    int i = blockIdx.x * blockDim.x + threadIdx.x;
    if (i < n) p[i] = 0.0f;
}

__global__ void k_pool(const float* __restrict__ h, const int* __restrict__ batch,
                       float* __restrict__ sums, float* __restrict__ cnts, int n) {
    int i = blockIdx.x * blockDim.x + threadIdx.x;
    if (i >= n) return;
    int g = batch[i];
    const float* hr = h + (size_t)i * HID;
    float* s = sums + (size_t)g * HID;
#pragma unroll
    for (int j = 0; j < HID; ++j) atomAddF(s + j, hr[j]);
    atomAddF(cnts + g, 1.0f);
}

__global__ void k_final(const float* __restrict__ sums, const float* __restrict__ cnts,
                        const float* __restrict__ Wo, const float* __restrict__ bo,
                        float* __restrict__ out, int G) {
    int g = blockIdx.x * blockDim.x + threadIdx.x;
    if (g >= G) return;
    float inv = 1.0f / fmaxf(cnts[g], 1.0f);
    float acc = bo[0];
#pragma unroll
    for (int j = 0; j < HID; ++j) acc += sums[g * HID + j] * inv * Wo[j];
    out[g] = acc;
}

// ---------------- host-side orchestration ----------------

extern "C" void kernel_launch(void* const* d_in, const int* in_sizes, int n_in,
                              void* d_out, int out_size, void* d_ws, size_t ws_size,
                              hipStream_t stream) {
    const float* x     = (const float*)d_in[0];
    const int*   ei    = (const int*)  d_in[1];
    const int*   batch = (const int*)  d_in[2];
    const float* ew    = (const float*)d_in[3];
    const float* W1    = (const float*)d_in[4];
    const float* b1    = (const float*)d_in[5];
    const float* Wh    = (const float*)d_in[6];
    const float* bh    = (const float*)d_in[7];
    const float* Wo    = (const float*)d_in[8];
    const float* bo    = (const float*)d_in[9];
    float* out = (float*)d_out;

    const int N = in_sizes[2];          // 200000 nodes
    const int E = in_sizes[3];          // 6400000 edges
    const int G = out_size;             // 1024 graphs
    const int* row = ei;
    const int* col = ei + E;

    float* ws   = (float*)d_ws;
    float* deg  = ws;                ws += N;
    float* dinv = ws;                ws += N;
    float* nrm  = ws;                ws += E;
    float* hw   = ws;                ws += (size_t)N * HID;
    float* hbuf = ws;                ws += (size_t)N * HID;
    float* agg  = ws;                ws += (size_t)N * HID;
    float* sums = ws;                ws += (size_t)G * HID;
    float* cnts = ws;                // G floats, contiguous after sums

    auto cdiv = [](int a, int b) { return (a + b - 1) / b; };
    const int T = 256;

    // degree + per-edge normalization
    k_init_deg<<<cdiv(N, T), T, 0, stream>>>(deg, N);
    k_deg_acc <<<cdiv(E, T), T, 0, stream>>>(col, ew, deg, E);
    k_norm    <<<cdiv(E, T), T, 0, stream>>>(row, col, ew, deg, nrm, E);
    k_deginv  <<<cdiv(N, T), T, 0, stream>>>(deg, dinv, N);

    const int ntiles  = N / 16;                 // 200000 % 16 == 0
    const int gblocks = cdiv(ntiles, 8);        // 8 waves per 256-thread block
    const int eblocks = cdiv(E, T);
    const int pblocks = cdiv(N * HID, T);

    // layer 1 (K=4 input)
    k_gemm1<<<gblocks, T, 0, stream>>>(x, W1, b1, dinv, hw, agg, ntiles);
    k_edge <<<eblocks, T, 0, stream>>>(row, col, nrm, hw, agg, E);
    k_post <<<pblocks, T, 0, stream>>>(hbuf, agg, N * HID, 0);

    // layers 2..8
    for (int k = 0; k < 7; ++k) {
        k_gemm16<<<gblocks, T, 0, stream>>>(hbuf, Wh + (size_t)k * HID * HID,
                                            bh + (size_t)k * HID, dinv, hw, agg, ntiles);
        k_edge  <<<eblocks, T, 0, stream>>>(row, col, nrm, hw, agg, E);
        k_post  <<<pblocks, T, 0, stream>>>(hbuf, agg, N * HID, (k < 6) ? 1 : 2);
    }

    // global mean pool + head
    k_zero <<<cdiv(G * (HID + 1), T), T, 0, stream>>>(sums, G * (HID + 1));
    k_pool <<<cdiv(N, T), T, 0, stream>>>(hbuf, batch, sums, cnts, N);
    k_final<<<cdiv(G, T), T, 0, stream>>>(sums, cnts, Wo, bo, out, G);
}